// FPSFunction_38001870635077
// MI455X (gfx1250) — compile-verified
//
#include <hip/hip_runtime.h>
#include <stdint.h>

// Farthest Point Sampling for MI455X (gfx1250), B=8, P=131072, nsamples=1024.
//
//  - 32 blocks per batch, 512 threads (16 wave32); each block owns a 4096-point
//    span staged ONCE into LDS (SoA, 48KB) via CDNA5 async global->LDS DMA
//    (global_load_async_to_lds_b128 + s_wait_asynccnt).
//  - Running min-distance lives in 8 VGPRs/lane; each of the 1024 sequential
//    rounds is LDS+VGPR work plus one L2-atomic inter-block combine per batch.
//  - Argmax via packed 64-bit key (f32 bits << 32 | ~idx): max-key == (max
//    value, min index), matching jnp.argmax first-index tie-break.
//  - Winner coords travel with the candidate record: zero global point loads
//    inside the loop. Only 2 workgroup barriers per round (publisher/spinner
//    and candidate readers are the same wave32, so no barrier between them).

#define B_      8
#define BPB     32          // blocks per batch
#define SPAN    4096        // points per block (P / BPB)
#define THREADS 512
#define PPT     8           // SPAN / THREADS
#define CHUNK   1024        // staging chunk (points)
#define NCHUNK  (SPAN / CHUNK)
#define INF_F   1e10f

__device__ __forceinline__ unsigned long long fps_pack(float v, int idx) {
  // v >= 0 always (squared distance or +INF init) -> float bits are
  // order-preserving as unsigned. ~idx makes ties prefer the smaller index.
  return ((unsigned long long)__float_as_uint(v) << 32) |
         (unsigned long long)(unsigned)(~(unsigned)idx);
}

__global__ __launch_bounds__(THREADS)
void fps_kernel(const float* __restrict__ pts,
                int* __restrict__ oIdx, float* __restrict__ oXyz,
                unsigned* __restrict__ cnt,
                unsigned long long* __restrict__ ckey,
                float* __restrict__ cx, float* __restrict__ cy,
                float* __restrict__ cz,
                int ns, int P)
{
  __shared__ float sx[SPAN];
  __shared__ float sy[SPAN];
  __shared__ float sz[SPAN];
  __shared__ __align__(16) float stage[CHUNK * 3];
  __shared__ unsigned long long wkey[THREADS / 32];
  __shared__ int   wsel;
  __shared__ float wxs, wys, wzs;

  const int tid  = threadIdx.x;
  const int blk  = blockIdx.x;          // 0..BPB-1
  const int b    = blockIdx.y;          // 0..B-1
  const int base = blk * SPAN;
  const float* bp = pts + (size_t)b * (size_t)P * 3;

  // ---------- one-time: async-DMA this block's span into LDS, SoA ----------
  {
    const unsigned stageOff = (unsigned)(uintptr_t)&stage[0]; // LDS byte offset
    for (int c = 0; c < NCHUNK; ++c) {
      const unsigned long long src =
          (unsigned long long)(uintptr_t)(bp + (size_t)(base + c * CHUNK) * 3);
      const int ncopies = (CHUNK * 12) / 16;          // 16B per lane-copy
      for (int k = tid; k < ncopies; k += THREADS) {
        unsigned dOff = stageOff + (unsigned)(k * 16);
        unsigned long long g = src + (unsigned long long)(k * 16);
        asm volatile("global_load_async_to_lds_b128 %0, %1, off"
                     :: "v"(dOff), "v"(g) : "memory");
      }
      asm volatile("s_wait_asynccnt 0" ::: "memory");
      __syncthreads();
      for (int p = tid; p < CHUNK; p += THREADS) {
        int li = c * CHUNK + p;
        sx[li] = stage[p * 3 + 0];
        sy[li] = stage[p * 3 + 1];
        sz[li] = stage[p * 3 + 2];
      }
      __syncthreads();
    }
  }

  float mind[PPT];
  #pragma unroll
  for (int j = 0; j < PPT; ++j) mind[j] = INF_F;

  int   last = 0;
  float lx = bp[0], ly = bp[1], lz = bp[2];   // coords of point 0 (uniform)

  const unsigned cntOff  = (unsigned)(b * 32);       // counter slot (128B stride)
  const int      slotBlk = b * BPB;                  // candidate row for batch b

  for (int t = 0; t < ns; ++t) {
    if (blk == 0 && tid == 0) {
      oIdx[(size_t)b * ns + t] = last;
      float* o = oXyz + ((size_t)b * ns + t) * 3;
      o[0] = lx; o[1] = ly; o[2] = lz;
    }

    // ---- update running min distance; lane-local argmax over PPT points ----
    unsigned long long key = 0ull;   // identity: below any real key
    #pragma unroll
    for (int j = 0; j < PPT; ++j) {
      int li = j * THREADS + tid;
      float dx = sx[li] - lx;
      float dy = sy[li] - ly;
      float dz = sz[li] - lz;
      float d  = dx * dx + dy * dy + dz * dz;
      float m  = fminf(mind[j], d);
      mind[j]  = m;
      unsigned long long k2 = fps_pack(m, base + li);
      key = (k2 > key) ? k2 : key;
    }

    // ---- wave32 reduce ----
    #pragma unroll
    for (int o = 16; o > 0; o >>= 1) {
      unsigned long long ok = __shfl_xor(key, o, 32);
      key = (ok > key) ? ok : key;
    }
    if ((tid & 31) == 0) wkey[tid >> 5] = key;
    __syncthreads();                                   // (A) wkey ready

    // ---- wave 0 does everything else: cross-wave reduce, publish, spin,
    //      candidate combine. Lane 0's spin and lanes 0..31's candidate reads
    //      are in the SAME wave -> program-ordered, no workgroup barrier.
    if (tid < 32) {
      const int rowBase = ((t & 1) * B_ * BPB) + slotBlk;

      unsigned long long k = (tid < (THREADS / 32)) ? wkey[tid] : 0ull;
      #pragma unroll
      for (int o = 16; o > 0; o >>= 1) {
        unsigned long long ok = __shfl_xor(k, o, 32);
        k = (ok > k) ? ok : k;
      }
      if (tid == 0) {
        int bi = (int)(~(unsigned)(k & 0xffffffffull));   // block-best index
        int li = bi - base;
        int slot = rowBase + blk;                          // double-buffered
        ckey[slot] = k;
        cx[slot] = sx[li]; cy[slot] = sy[li]; cz[slot] = sz[li];
        __threadfence();                                   // release
        atomicAdd(&cnt[cntOff], 1u);
        unsigned tgt = (unsigned)(BPB * (t + 1));
        while (__hip_atomic_load(&cnt[cntOff], __ATOMIC_ACQUIRE,
                                 __HIP_MEMORY_SCOPE_AGENT) < tgt)
          __builtin_amdgcn_s_sleep(1);
      }

      // combine the 32 block candidates (identical result in every block)
      int slot = rowBase + tid;
      unsigned long long ck =
          __hip_atomic_load(&ckey[slot], __ATOMIC_RELAXED, __HIP_MEMORY_SCOPE_AGENT);
      float x = __hip_atomic_load(&cx[slot], __ATOMIC_RELAXED, __HIP_MEMORY_SCOPE_AGENT);
      float y = __hip_atomic_load(&cy[slot], __ATOMIC_RELAXED, __HIP_MEMORY_SCOPE_AGENT);
      float z = __hip_atomic_load(&cz[slot], __ATOMIC_RELAXED, __HIP_MEMORY_SCOPE_AGENT);
      #pragma unroll
      for (int o = 16; o > 0; o >>= 1) {
        unsigned long long ok = __shfl_xor(ck, o, 32);
        float ox = __shfl_xor(x, o, 32);
        float oy = __shfl_xor(y, o, 32);
        float oz = __shfl_xor(z, o, 32);
        if (ok > ck) { ck = ok; x = ox; y = oy; z = oz; }
      }
      if (tid == 0) {
        wsel = (int)(~(unsigned)(ck & 0xffffffffull));
        wxs = x; wys = y; wzs = z;
      }
    }
    __syncthreads();                                   // (C) winner ready
    last = wsel; lx = wxs; ly = wys; lz = wzs;
  }
}

extern "C" void kernel_launch(void* const* d_in, const int* in_sizes, int n_in,
                              void* d_out, int out_size, void* d_ws, size_t ws_size,
                              hipStream_t stream) {
  (void)n_in; (void)ws_size;
  const float* pts = (const float*)d_in[0];
  const int P  = in_sizes[0] / (B_ * 3);     // 131072
  const int ns = out_size / (B_ * 4);        // B*ns ints + B*ns*3 floats = 4*B*ns

  int*   oIdx = (int*)d_out;                               // first B*ns int32
  float* oXyz = (float*)d_out + (size_t)B_ * ns;           // then B*ns*3 f32

  // workspace layout: [0,1024) counters (8 batches, 128B stride),
  // [1024,5120) ckey (2*8*32 u64), then cx/cy/cz (2*8*32 f32 each).
  unsigned*           cnt  = (unsigned*)d_ws;
  unsigned long long* ckey = (unsigned long long*)((char*)d_ws + 1024);
  float*              cx   = (float*)((char*)d_ws + 1024 + 4096);
  float*              cy   = cx + 2 * B_ * BPB;
  float*              cz   = cy + 2 * B_ * BPB;

  hipMemsetAsync(d_ws, 0, 1024, stream);     // reset barrier counters (capture-legal)

  dim3 grid(BPB, B_);
  dim3 block(THREADS);
  hipLaunchKernelGGL(fps_kernel, grid, block, 0, stream,
                     pts, oIdx, oXyz, cnt, ckey, cx, cy, cz, ns, P);
}